// corr_hzp_893353197596
// MI455X (gfx1250) — compile-verified
//
#include <hip/hip_runtime.h>

// Local correlation volume (radius 4, 81 displacement channels) on gfx1250.
// Per (b, h, 16-wide w-tile): correlation as 16x64 @ 64x24 GEMMs (one per
// vertical shift i) via V_WMMA_F32_16X16X4_F32; the 9 horizontal shifts are
// the diagonal bands D[m, m+j]. Diagonals are staged through LDS (branch-free
// scatter with a dump slot) and written back as coalesced 64B row segments.

#define RADIUS 4
#define NCH    64
#define HH     256
#define WW     256
#define NB     4
#define ND     81      // (2*RADIUS+1)^2
#define TW     16      // pixels per wave (WMMA M)
#define THR    8       // h-rows per workgroup (one wave each)

#define SROW   17      // LDS row stride (16 + 1 pad -> conflict-free scatter)
#define SDUMP  156     // dump slot for invalid (m,j) combos
#define SLDS   160     // per-wave LDS floats

typedef __attribute__((ext_vector_type(2))) float v2f;
typedef __attribute__((ext_vector_type(8))) float v8f;

__device__ __forceinline__ int reflect256(int t) {
    // reflect-pad index into [0, 256) for t in [-4, 259]
    t = (t < 0) ? -t : t;
    t = (t >= 256) ? (510 - t) : t;
    return t;
}

__global__ __launch_bounds__(256) void corr_hzp_wmma_kernel(
    const float* __restrict__ x,
    const float* __restrict__ y,
    float* __restrict__ out)
{
    __shared__ float sout[THR][SLDS];    // 8 * 640B = 5KB, per-wave private

    const int lane = threadIdx.x & 31;   // wave32
    const int wave = threadIdx.x >> 5;

    const int wtiles = WW / TW;          // 16
    const int htiles = HH / THR;         // 32
    const int gid = blockIdx.x;
    const int b   = gid / (wtiles * htiles);
    const int h0  = ((gid / wtiles) % htiles) * THR;
    const int w0  = (gid % wtiles) * TW;
    const int h   = h0 + wave;

    const int half = lane >> 4;          // 0: K={0,1}, 1: K={2,3} per 4-wide k-step
    const int lm   = lane & 15;          // M (for A) / N (for B) index

    const size_t plane = (size_t)HH * WW;

    // ---- A fragments: x[b, c, h, w0+lm], 16 k-steps x 4 channels (register resident) ----
    v2f af[16];
    const float* xb = x + ((size_t)b * NCH) * plane + (size_t)h * WW + (w0 + lm);
#pragma unroll
    for (int t = 0; t < 16; ++t) {
        const int c0 = 4 * t + 2 * half;
        v2f a;
        a.x = xb[(size_t)c0       * plane];
        a.y = xb[(size_t)(c0 + 1) * plane];
        af[t] = a;
    }

    const float* ybase = y + ((size_t)b * NCH) * plane;
    float* outb = out + (((size_t)b * ND) * HH + h) * WW;

#pragma unroll 1
    for (int i = 0; i < 9; ++i) {                 // vertical displacement
        const int hy = reflect256(h + i - RADIUS);

        // prefetch next vertical row of y into cache (global_prefetch_b8)
        if (i < 8) {
            const int hyn = reflect256(h + i - RADIUS + 1);
            __builtin_prefetch(ybase + (size_t)hyn * WW + w0, 0, 0);
        }

#pragma unroll
        for (int tile = 0; tile < 2; ++tile) {    // two N-tiles covering n in [0,24)
            const int nbase = w0 - RADIUS + 8 * tile;
            const int col   = reflect256(nbase + lm);
            const float* yb = ybase + (size_t)hy * WW + col;

            v8f acc = {};
#pragma unroll
            for (int t = 0; t < 16; ++t) {        // K = 64 in 16 steps of 4
                const int c0 = 4 * t + 2 * half;
                v2f bf;
                bf.x = yb[(size_t)c0       * plane];
                bf.y = yb[(size_t)(c0 + 1) * plane];
                // D = A(16x4) x B(4x16) + C   -> v_wmma_f32_16x16x4_f32
                acc = __builtin_amdgcn_wmma_f32_16x16x4_f32(
                        false, af[t], false, bf, (short)0, acc, false, false);
            }

            // ---- branch-free diagonal scatter into LDS: sout[j*17 + m] = D[m, m+j] ----
            const int nglob  = lm + 8 * tile;                      // global n in [0,24)
            const bool owner = (tile == 0) ? (nglob < 16) : (nglob >= 16);
#pragma unroll
            for (int r = 0; r < 8; ++r) {
                const int m = r + 8 * half;                        // D row held in acc[r]
                const int j = nglob - m;                           // horizontal shift (0..8)
                const bool ok = owner && (j >= 0) && (j <= 8);
                const int addr = ok ? (j * SROW + m) : SDUMP;      // unconditional ds_store
                sout[wave][addr] = acc[r];
            }
        }

        // ---- coalesced write-back: 9 rows of 16 consecutive w per (i,j) ----
        // LDS ops are in-order within a wave, so the scatter above is visible.
#pragma unroll
        for (int k = 0; k < 5; ++k) {
            const int idx = k * 32 + lane;        // 0..159; valid < 144
            const int j   = idx >> 4;
            const int m   = idx & 15;
            const float v = sout[wave][j * SROW + m];
            if (idx < 144) {
                outb[((size_t)(i * 9 + j)) * plane + (w0 + m)] = v;
            }
        }
    }
}

extern "C" void kernel_launch(void* const* d_in, const int* in_sizes, int n_in,
                              void* d_out, int out_size, void* d_ws, size_t ws_size,
                              hipStream_t stream) {
    const float* x = (const float*)d_in[0];
    const float* y = (const float*)d_in[1];
    float* out = (float*)d_out;
    (void)in_sizes; (void)n_in; (void)out_size; (void)d_ws; (void)ws_size;

    const int blocks = NB * (HH / THR) * (WW / TW);   // 4 * 32 * 16 = 2048
    corr_hzp_wmma_kernel<<<blocks, 256, 0, stream>>>(x, y, out);
}